// SpatialGNN_72653666779577
// MI455X (gfx1250) — compile-verified
//
#include <hip/hip_runtime.h>
#include <math.h>

#define HEADS 4
#define HD 128
#define NEG_SLOPE 0.2f

typedef __attribute__((ext_vector_type(2))) float v2f;
typedef __attribute__((ext_vector_type(8))) float v8f;

__device__ __forceinline__ float atomicMaxF(float* addr, float val) {
    // IEEE-754 ordering trick: for >=0 use signed int max, for <0 use unsigned min
    if (val >= 0.0f) {
        return __int_as_float(atomicMax((int*)addr, __float_as_int(val)));
    } else {
        return __uint_as_float(atomicMin((unsigned int*)addr, __float_as_uint(val)));
    }
}

// ---------------------------------------------------------------------------
// Fused dual GEMM: h = x @ gw ; y = relu(bn(x @ fw + fb))
// Block = 256 threads (8 waves). Block covers 16 rows; wave w covers cols
// [16w, 16w+16) of the 128 output columns, accumulating BOTH weight matrices.
// Uses V_WMMA_F32_16X16X4_F32 (full fp32 precision on the matrix pipes).
// ---------------------------------------------------------------------------
__global__ __launch_bounds__(256) void gemm_dual_kernel(
    const float* __restrict__ x, int din,
    const float* __restrict__ gw, const float* __restrict__ fw,
    const float* __restrict__ fb, const float* __restrict__ fg,
    const float* __restrict__ fbeta,
    float* __restrict__ h, float* __restrict__ y)
{
    __shared__ float lds_x[16][68];   // pad 64->68 to spread banks

    const int tid  = threadIdx.x;
    const int wave = tid >> 5;
    const int lane = tid & 31;
    const int half = lane >> 4;       // 0: lanes 0-15, 1: lanes 16-31
    const int l    = lane & 15;
    const int r0   = blockIdx.x * 16;
    const int c0   = wave * 16;

    v8f acc_h = {};
    v8f acc_y = {};

    for (int k0 = 0; k0 < din; k0 += 64) {
        __syncthreads();
        {   // cooperative 16x64 tile load: 4 floats per thread (float4)
            int idx = tid * 4;
            int row = idx >> 6;
            int col = idx & 63;
            const float4 v = *(const float4*)(x + (size_t)(r0 + row) * din + k0 + col);
            lds_x[row][col + 0] = v.x;
            lds_x[row][col + 1] = v.y;
            lds_x[row][col + 2] = v.z;
            lds_x[row][col + 3] = v.w;
        }
        __syncthreads();

        #pragma unroll
        for (int kk = 0; kk < 64; kk += 4) {
            // A (16x4 f32): lane half selects K pair {0,1} vs {2,3}
            v2f a;
            a.x = lds_x[l][kk + half * 2 + 0];
            a.y = lds_x[l][kk + half * 2 + 1];
            // B (4x16 f32): row-major weights, row k contiguous over 128 cols
            const int kg = k0 + kk + half * 2;
            v2f bg, bf;
            bg.x = gw[(size_t)(kg + 0) * HD + c0 + l];
            bg.y = gw[(size_t)(kg + 1) * HD + c0 + l];
            bf.x = fw[(size_t)(kg + 0) * HD + c0 + l];
            bf.y = fw[(size_t)(kg + 1) * HD + c0 + l];

            acc_h = __builtin_amdgcn_wmma_f32_16x16x4_f32(
                false, a, false, bg, (short)0, acc_h, false, false);
            acc_y = __builtin_amdgcn_wmma_f32_16x16x4_f32(
                false, a, false, bf, (short)0, acc_y, false, false);
        }
    }

    // D layout: VGPR r -> row (half*8 + r), col = c0 + l
    const int   col   = c0 + l;
    const float scale = fg[col] * rsqrtf(1.0f + 1e-5f);
    const float bias  = fb[col];
    const float beta  = fbeta[col];
    #pragma unroll
    for (int r = 0; r < 8; ++r) {
        const int row = r0 + half * 8 + r;
        h[(size_t)row * HD + col] = acc_h[r];
        float v = (acc_y[r] + bias) * scale + beta;
        y[(size_t)row * HD + col] = fmaxf(v, 0.0f);
    }
}

// al_s[n,h] = sum_c h[n,h,c]*a_src[h,c] ; al_d likewise
__global__ void attn_logits_kernel(const float* __restrict__ h,
                                   const float* __restrict__ gas,
                                   const float* __restrict__ gad,
                                   float* __restrict__ al_s,
                                   float* __restrict__ al_d, int N)
{
    int gid = blockIdx.x * blockDim.x + threadIdx.x;
    if (gid >= N * HEADS) return;
    int n  = gid >> 2;
    int hh = gid & 3;
    const float* hp = h   + (size_t)n * HD + hh * 32;
    const float* as = gas + hh * 32;
    const float* ad = gad + hh * 32;
    float ss = 0.0f, sd = 0.0f;
    #pragma unroll
    for (int c = 0; c < 32; ++c) {
        float hv = hp[c];
        ss += hv * as[c];
        sd += hv * ad[c];
    }
    al_s[gid] = ss;
    al_d[gid] = sd;
}

__global__ void init_kernel(float* __restrict__ m, float* __restrict__ s,
                            float* __restrict__ accum, int N)
{
    int gid = blockIdx.x * blockDim.x + threadIdx.x;
    if (gid < N * HD) accum[gid] = 0.0f;
    if (gid < N * HEADS) { m[gid] = -INFINITY; s[gid] = 0.0f; }
}

// pass 1: e = leaky_relu(al_s[src]+al_d[dst]); m[dst] = max(m[dst], e)
__global__ void edge_max_kernel(const int* __restrict__ ei, int E, int N,
                                const float* __restrict__ al_s,
                                const float* __restrict__ al_d,
                                float* __restrict__ ebuf, float* __restrict__ m)
{
    int gid = blockIdx.x * blockDim.x + threadIdx.x;
    int total = (E + N) * HEADS;
    if (gid >= total) return;
    int e = gid >> 2, hh = gid & 3;
    int src, dst;
    if (e < E) { src = ei[e]; dst = ei[E + e]; }
    else       { src = dst = e - E; }              // self loop
    float v = al_s[src * 4 + hh] + al_d[dst * 4 + hh];
    v = v > 0.0f ? v : v * NEG_SLOPE;
    ebuf[gid] = v;
    atomicMaxF(&m[dst * 4 + hh], v);
}

// pass 2: p = exp(e - m[dst]); s[dst] += p  (p stored in-place in ebuf)
__global__ void edge_sum_kernel(const int* __restrict__ ei, int E, int N,
                                const float* __restrict__ m,
                                float* __restrict__ ebuf, float* __restrict__ s)
{
    int gid = blockIdx.x * blockDim.x + threadIdx.x;
    int total = (E + N) * HEADS;
    if (gid >= total) return;
    int e = gid >> 2, hh = gid & 3;
    int dst = (e < E) ? ei[E + e] : (e - E);
    float p = __expf(ebuf[gid] - m[dst * 4 + hh]);
    ebuf[gid] = p;
    atomicAdd(&s[dst * 4 + hh], p);
}

// pass 3: accum[dst] += (p/s[dst]) * h[src].  32 lanes per edge, float4/lane.
__global__ void edge_scatter_kernel(const int* __restrict__ ei, int E, int N,
                                    const float* __restrict__ h,
                                    const float* __restrict__ ebuf,
                                    const float* __restrict__ s,
                                    float* __restrict__ accum)
{
    long long gid = (long long)blockIdx.x * blockDim.x + threadIdx.x;
    long long total = (long long)(E + N) * 32;
    if (gid >= total) return;
    int e    = (int)(gid >> 5);
    int lane = (int)(gid & 31);
    int src, dst;
    if (e < E) { src = ei[e]; dst = ei[E + e]; }
    else       { src = dst = e - E; }
    int hh = lane >> 3;                            // feature lane*4 -> head
    float alpha = ebuf[e * 4 + hh] / s[dst * 4 + hh];
    int f = lane * 4;
    const float4 hv = *(const float4*)(h + (size_t)src * HD + f);
    float* ap = accum + (size_t)dst * HD + f;
    atomicAdd(ap + 0, alpha * hv.x);
    atomicAdd(ap + 1, alpha * hv.y);
    atomicAdd(ap + 2, alpha * hv.z);
    atomicAdd(ap + 3, alpha * hv.w);
}

// x_next = relu(gat_accum + gb + y)
__global__ void finalize_kernel(const float* __restrict__ accum,
                                const float* __restrict__ y,
                                const float* __restrict__ gb,
                                float* __restrict__ xnext, int N)
{
    int gid = blockIdx.x * blockDim.x + threadIdx.x;
    if (gid >= N * HD) return;
    int col = gid & (HD - 1);
    float v = accum[gid] + gb[col] + y[gid];
    xnext[gid] = fmaxf(v, 0.0f);
}

__global__ void classifier_kernel(const float* __restrict__ x,
                                  const float* __restrict__ w,
                                  const float* __restrict__ b,
                                  float* __restrict__ out, int N)
{
    int gid = blockIdx.x * blockDim.x + threadIdx.x;
    if (gid >= N * 10) return;
    int n = gid / 10, o = gid % 10;
    float acc = b[o];
    #pragma unroll 8
    for (int c = 0; c < HD; ++c) acc += x[(size_t)n * HD + c] * w[c * 10 + o];
    out[gid] = acc;
}

extern "C" void kernel_launch(void* const* d_in, const int* in_sizes, int n_in,
                              void* d_out, int out_size, void* d_ws, size_t ws_size,
                              hipStream_t stream) {
    const float* x_in = (const float*)d_in[0];
    const int*   ei   = (const int*)d_in[1];
    const int D = 512;
    const int N = in_sizes[0] / D;      // 80000
    const int E = in_sizes[1] / 2;      // 1000000

    // workspace carve-out (256B aligned slices)
    char* ws = (char*)d_ws;
    size_t off = 0;
    auto carve = [&](size_t bytes) -> void* {
        void* p = ws + off;
        off = (off + bytes + 255) & ~(size_t)255;
        return p;
    };
    float* h     = (float*)carve((size_t)N * HD * 4);
    float* y     = (float*)carve((size_t)N * HD * 4);
    float* accum = (float*)carve((size_t)N * HD * 4);
    float* xbuf  = (float*)carve((size_t)N * HD * 4);
    float* al_s  = (float*)carve((size_t)N * HEADS * 4);
    float* al_d  = (float*)carve((size_t)N * HEADS * 4);
    float* m     = (float*)carve((size_t)N * HEADS * 4);
    float* s     = (float*)carve((size_t)N * HEADS * 4);
    float* ebuf  = (float*)carve((size_t)(E + N) * HEADS * 4);
    (void)ws_size; (void)n_in; (void)out_size;

    const int base[3] = {3, 11, 19};    // l{0,1,2}_gw index in d_in
    for (int l = 0; l < 3; ++l) {
        const float* gw    = (const float*)d_in[base[l] + 0];
        const float* gas   = (const float*)d_in[base[l] + 1];
        const float* gad   = (const float*)d_in[base[l] + 2];
        const float* gb    = (const float*)d_in[base[l] + 3];
        const float* fw    = (const float*)d_in[base[l] + 4];
        const float* fb    = (const float*)d_in[base[l] + 5];
        const float* fg    = (const float*)d_in[base[l] + 6];
        const float* fbeta = (const float*)d_in[base[l] + 7];
        const float* xin   = (l == 0) ? x_in : xbuf;
        const int    din   = (l == 0) ? D : HD;

        gemm_dual_kernel<<<N / 16, 256, 0, stream>>>(xin, din, gw, fw, fb, fg,
                                                     fbeta, h, y);
        attn_logits_kernel<<<(N * HEADS + 255) / 256, 256, 0, stream>>>(
            h, gas, gad, al_s, al_d, N);
        init_kernel<<<(N * HD + 255) / 256, 256, 0, stream>>>(m, s, accum, N);

        const int et4 = (E + N) * HEADS;
        edge_max_kernel<<<(et4 + 255) / 256, 256, 0, stream>>>(ei, E, N, al_s,
                                                               al_d, ebuf, m);
        edge_sum_kernel<<<(et4 + 255) / 256, 256, 0, stream>>>(ei, E, N, m,
                                                               ebuf, s);
        const long long tot = (long long)(E + N) * 32;
        edge_scatter_kernel<<<(int)((tot + 255) / 256), 256, 0, stream>>>(
            ei, E, N, h, ebuf, s, accum);
        finalize_kernel<<<(N * HD + 255) / 256, 256, 0, stream>>>(accum, y, gb,
                                                                  xbuf, N);
    }

    classifier_kernel<<<(N * 10 + 255) / 256, 256, 0, stream>>>(
        xbuf, (const float*)d_in[27], (const float*)d_in[28], (float*)d_out, N);
}